// PoseDecoder_31490700214873
// MI455X (gfx1250) — compile-verified
//
#include <hip/hip_runtime.h>
#include <hip/hip_bf16.h>
#include <stdint.h>

#define BLOCK 128
#define RWORDS 102   // 17 joints * 6 floats
#define OWORDS 51    // 17 joints * 3 floats
#define ROT_BYTES (BLOCK * RWORDS * 4)          // 52224
#define ROOT_LDS_OFF ROT_BYTES                  // root region starts here
#define LDS_FLOATS (BLOCK * RWORDS + BLOCK * 3) // 13056 + 384 = 13440 floats

struct V3 { float x, y, z; };
struct M3 { V3 cx, cy, cz; };   // columns (x, y, z basis vectors)

__device__ __forceinline__ V3 cross3(V3 a, V3 b) {
    V3 r;
    r.x = a.y * b.z - a.z * b.y;
    r.y = a.z * b.x - a.x * b.z;
    r.z = a.x * b.y - a.y * b.x;
    return r;
}

__device__ __forceinline__ V3 norm3(V3 v) {
    float n = sqrtf(v.x * v.x + v.y * v.y + v.z * v.z);
    float r = 1.0f / fmaxf(n, 1e-8f);
    V3 o; o.x = v.x * r; o.y = v.y * r; o.z = v.z * r;
    return o;
}

__device__ __forceinline__ M3 rot6d(const float* r) {
    V3 xr; xr.x = r[0]; xr.y = r[1]; xr.z = r[2];
    V3 yr; yr.x = r[3]; yr.y = r[4]; yr.z = r[5];
    V3 x = norm3(xr);
    V3 z = norm3(cross3(x, yr));
    V3 y = cross3(z, x);
    M3 m; m.cx = x; m.cy = y; m.cz = z;
    return m;
}

__device__ __forceinline__ V3 mv(const M3& A, float b0, float b1, float b2) {
    V3 r;
    r.x = A.cx.x * b0 + A.cy.x * b1 + A.cz.x * b2;
    r.y = A.cx.y * b0 + A.cy.y * b1 + A.cz.y * b2;
    r.z = A.cx.z * b0 + A.cy.z * b1 + A.cz.z * b2;
    return r;
}

__device__ __forceinline__ M3 mm(const M3& A, const M3& B) {
    M3 C;
    C.cx = mv(A, B.cx.x, B.cx.y, B.cx.z);
    C.cy = mv(A, B.cy.x, B.cy.y, B.cy.z);
    C.cz = mv(A, B.cz.x, B.cz.y, B.cz.z);
    return C;
}

__device__ __forceinline__ V3 axpy(V3 p, float s, V3 c) {
    V3 r; r.x = p.x + s * c.x; r.y = p.y + s * c.y; r.z = p.z + s * c.z;
    return r;
}

// Full forward kinematics for one sample (leaf-joint rotations elided).
__device__ __forceinline__ void decode_pose(const float* r6, const float* len,
                                            V3 p0, float* P) {
    auto st = [&](int j, V3 v) { P[3*j+0] = v.x; P[3*j+1] = v.y; P[3*j+2] = v.z; };

    M3 Rw0 = rot6d(r6 + 0);
    st(0, p0);

    // chain 0 -> 1 -> 2 -> 3   (ops: -x, -y, -y)
    M3 Ra = mm(Rw0, rot6d(r6 + 6));          // Rw1
    V3 p1 = axpy(p0, -len[0], Rw0.cx); st(1, p1);
    M3 Rb = mm(Ra, rot6d(r6 + 12));          // Rw2
    V3 p2 = axpy(p1, -len[1], Ra.cy);  st(2, p2);
    st(3, axpy(p2, -len[2], Rb.cy));

    // chain 0 -> 4 -> 5 -> 6   (ops: +x, -y, -y)
    Ra = mm(Rw0, rot6d(r6 + 24));            // Rw4
    V3 p4 = axpy(p0,  len[3], Rw0.cx); st(4, p4);
    Rb = mm(Ra, rot6d(r6 + 30));             // Rw5
    V3 p5 = axpy(p4, -len[4], Ra.cy);  st(5, p5);
    st(6, axpy(p5, -len[5], Rb.cy));

    // spine 0 -> 7 -> 8        (ops: +y, +y)
    M3 Rw7 = mm(Rw0, rot6d(r6 + 42));
    V3 p7 = axpy(p0, len[6], Rw0.cy);  st(7, p7);
    M3 Rw8 = mm(Rw7, rot6d(r6 + 48));
    V3 p8 = axpy(p7, len[7], Rw7.cy);  st(8, p8);

    // head 8 -> 9 -> 10        (ops: +y, +y)
    Ra = mm(Rw8, rot6d(r6 + 54));            // Rw9
    V3 p9 = axpy(p8, len[8], Rw8.cy);  st(9, p9);
    st(10, axpy(p9, len[9], Ra.cy));

    // arm 8 -> 11 -> 12 -> 13  (ops: +x, +x, +x)
    Ra = mm(Rw8, rot6d(r6 + 66));            // Rw11
    V3 p11 = axpy(p8, len[10], Rw8.cx); st(11, p11);
    Rb = mm(Ra, rot6d(r6 + 72));             // Rw12
    V3 p12 = axpy(p11, len[11], Ra.cx); st(12, p12);
    st(13, axpy(p12, len[12], Rb.cx));

    // arm 8 -> 14 -> 15 -> 16  (ops: -x, -x, -x)
    Ra = mm(Rw8, rot6d(r6 + 84));            // Rw14
    V3 p14 = axpy(p8, -len[13], Rw8.cx); st(14, p14);
    Rb = mm(Ra, rot6d(r6 + 90));             // Rw15
    V3 p15 = axpy(p14, -len[14], Ra.cx); st(15, p15);
    st(16, axpy(p15, -len[15], Rb.cx));
}

// ---- CDNA5 async global<->LDS, single base VGPRs + immediate offsets ------
// INST_OFFSET is added to BOTH the global and the LDS address (ISA 08 §4.4),
// so symmetric staging needs only one address pair per transfer width.

#define A_LD128(LOFF, GPTR, OFF) \
    asm volatile("global_load_async_to_lds_b128 %0, %1, off offset:" OFF \
                 :: "v"(LOFF), "v"(GPTR) : "memory")
#define A_LD64(LOFF, GPTR, OFF) \
    asm volatile("global_load_async_to_lds_b64 %0, %1, off offset:" OFF \
                 :: "v"(LOFF), "v"(GPTR) : "memory")
#define A_LD32(LOFF, GPTR, OFF) \
    asm volatile("global_load_async_to_lds_b32 %0, %1, off offset:" OFF \
                 :: "v"(LOFF), "v"(GPTR) : "memory")
#define A_ST128(GPTR, LOFF, OFF) \
    asm volatile("global_store_async_from_lds_b128 %0, %1, off offset:" OFF \
                 :: "v"(GPTR), "v"(LOFF) : "memory")
#define A_ST64(GPTR, LOFF, OFF) \
    asm volatile("global_store_async_from_lds_b64 %0, %1, off offset:" OFF \
                 :: "v"(GPTR), "v"(LOFF) : "memory")
#define A_ST32(GPTR, LOFF, OFF) \
    asm volatile("global_store_async_from_lds_b32 %0, %1, off offset:" OFF \
                 :: "v"(GPTR), "v"(LOFF) : "memory")

__device__ __forceinline__ void wait_async0() {
    asm volatile("s_wait_asynccnt 0" ::: "memory");
}

// ---------------- main kernel: full blocks of 128 samples ------------------

__global__ void __launch_bounds__(BLOCK)
pose_main(const float* __restrict__ rot, const float* __restrict__ bones,
          const float* __restrict__ root, float* __restrict__ out, int L) {
    __shared__ float lds[LDS_FLOATS];   // 53760 bytes

    const int tid = threadIdx.x;
    const long long sample0 = (long long)blockIdx.x * BLOCK;
    const long long sample  = sample0 + tid;

    const uint32_t lbase = (uint32_t)(uintptr_t)&lds[0];

    // --- Stage rotations: 52224 B via 25 x b128 + 1 x b64, shared bases ---
    {
        const char* g16 = (const char*)(rot + (size_t)sample0 * RWORDS) + tid * 16;
        const uint32_t l16 = lbase + (uint32_t)(tid * 16);
        A_LD128(l16, g16, "0");     A_LD128(l16, g16, "2048");
        A_LD128(l16, g16, "4096");  A_LD128(l16, g16, "6144");
        A_LD128(l16, g16, "8192");  A_LD128(l16, g16, "10240");
        A_LD128(l16, g16, "12288"); A_LD128(l16, g16, "14336");
        A_LD128(l16, g16, "16384"); A_LD128(l16, g16, "18432");
        A_LD128(l16, g16, "20480"); A_LD128(l16, g16, "22528");
        A_LD128(l16, g16, "24576"); A_LD128(l16, g16, "26624");
        A_LD128(l16, g16, "28672"); A_LD128(l16, g16, "30720");
        A_LD128(l16, g16, "32768"); A_LD128(l16, g16, "34816");
        A_LD128(l16, g16, "36864"); A_LD128(l16, g16, "38912");
        A_LD128(l16, g16, "40960"); A_LD128(l16, g16, "43008");
        A_LD128(l16, g16, "45056"); A_LD128(l16, g16, "47104");
        A_LD128(l16, g16, "49152");
        const char* g8 = (const char*)(rot + (size_t)sample0 * RWORDS) + tid * 8;
        const uint32_t l8 = lbase + (uint32_t)(tid * 8);
        A_LD64(l8, g8, "51200");
        // --- Stage roots: 1536 B at LDS offset 52224 (b64 + b32 passes) ---
        const char* gr8 = (const char*)(root + (size_t)sample0 * 3) + tid * 8;
        const uint32_t lr8 = lbase + (uint32_t)(ROOT_LDS_OFF + tid * 8);
        A_LD64(lr8, gr8, "0");
        const char* gr4 = (const char*)(root + (size_t)sample0 * 3) + 1024 + tid * 4;
        const uint32_t lr4 = lbase + (uint32_t)(ROOT_LDS_OFF + 1024 + tid * 4);
        A_LD32(lr4, gr4, "0");
        wait_async0();
    }
    __syncthreads();

    // --- Per-thread FK from LDS (strides 102 / 3 words: conflict-free) ---
    const float* r6 = &lds[tid * RWORDS];
    const int b = (int)(sample / L);
    const float4* bl4 = (const float4*)(bones + (size_t)b * 16);
    float len[16];
    {
        float4 a = bl4[0], c = bl4[1], d = bl4[2], e = bl4[3];
        len[0]=a.x;  len[1]=a.y;  len[2]=a.z;  len[3]=a.w;
        len[4]=c.x;  len[5]=c.y;  len[6]=c.z;  len[7]=c.w;
        len[8]=d.x;  len[9]=d.y;  len[10]=d.z; len[11]=d.w;
        len[12]=e.x; len[13]=e.y; len[14]=e.z; len[15]=e.w;
    }
    V3 p0;
    p0.x = lds[BLOCK * RWORDS + tid * 3 + 0];
    p0.y = lds[BLOCK * RWORDS + tid * 3 + 1];
    p0.z = lds[BLOCK * RWORDS + tid * 3 + 2];

    float P[OWORDS];
    decode_pose(r6, len, p0, P);

    // --- Stage results in LDS (reuse rot region), coalesced async store ---
    __syncthreads();
    {
        float* so = &lds[tid * OWORDS];    // stride 51 words: conflict-free
        #pragma unroll
        for (int k = 0; k < OWORDS; ++k) so[k] = P[k];
    }
    __syncthreads();
    {
        // total BLOCK*204 = 26112 B = 12*2048 + 1024 + 512
        char* g16 = (char*)(out + (size_t)sample0 * OWORDS) + tid * 16;
        const uint32_t l16 = lbase + (uint32_t)(tid * 16);
        A_ST128(g16, l16, "0");     A_ST128(g16, l16, "2048");
        A_ST128(g16, l16, "4096");  A_ST128(g16, l16, "6144");
        A_ST128(g16, l16, "8192");  A_ST128(g16, l16, "10240");
        A_ST128(g16, l16, "12288"); A_ST128(g16, l16, "14336");
        A_ST128(g16, l16, "16384"); A_ST128(g16, l16, "18432");
        A_ST128(g16, l16, "20480"); A_ST128(g16, l16, "22528");
        char* g8 = (char*)(out + (size_t)sample0 * OWORDS) + tid * 8;
        const uint32_t l8 = lbase + (uint32_t)(tid * 8);
        A_ST64(g8, l8, "24576");
        char* g4 = (char*)(out + (size_t)sample0 * OWORDS) + tid * 4;
        const uint32_t l4 = lbase + (uint32_t)(tid * 4);
        A_ST32(g4, l4, "25600");
        wait_async0();
    }
}

// ---------------- tail kernel: any leftover samples (direct path) ----------

__global__ void pose_tail(const float* __restrict__ rot,
                          const float* __restrict__ bones,
                          const float* __restrict__ root,
                          float* __restrict__ out, int L,
                          long long start, long long total) {
    long long sample = start + (long long)blockIdx.x * blockDim.x + threadIdx.x;
    if (sample >= total) return;
    float r6[RWORDS];
    #pragma unroll
    for (int i = 0; i < RWORDS; ++i) r6[i] = rot[(size_t)sample * RWORDS + i];
    const float* bl = bones + (size_t)(sample / L) * 16;
    float len[16];
    #pragma unroll
    for (int i = 0; i < 16; ++i) len[i] = bl[i];
    V3 p0;
    p0.x = root[(size_t)sample * 3 + 0];
    p0.y = root[(size_t)sample * 3 + 1];
    p0.z = root[(size_t)sample * 3 + 2];
    float P[OWORDS];
    decode_pose(r6, len, p0, P);
    #pragma unroll
    for (int k = 0; k < OWORDS; ++k) out[(size_t)sample * OWORDS + k] = P[k];
}

extern "C" void kernel_launch(void* const* d_in, const int* in_sizes, int n_in,
                              void* d_out, int out_size, void* d_ws, size_t ws_size,
                              hipStream_t stream) {
    const float* rot   = (const float*)d_in[0];   // [BL, 17, 6]
    const float* bones = (const float*)d_in[1];   // [B, 16]
    const float* root  = (const float*)d_in[2];   // [BL, 3]
    float* out = (float*)d_out;                   // [BL, 17, 3]

    const long long BL = (long long)in_sizes[0] / RWORDS;
    const int B = in_sizes[1] / 16;
    const int L = (int)(BL / (B > 0 ? B : 1));

    const long long fullBlocks = BL / BLOCK;
    if (fullBlocks > 0) {
        pose_main<<<(int)fullBlocks, BLOCK, 0, stream>>>(rot, bones, root, out, L);
    }
    const long long rem = BL - fullBlocks * BLOCK;
    if (rem > 0) {
        pose_tail<<<(int)((rem + 127) / 128), 128, 0, stream>>>(
            rot, bones, root, out, L, fullBlocks * BLOCK, BL);
    }
}